// SAModule_66649302499625
// MI455X (gfx1250) — compile-verified
//
#include <hip/hip_runtime.h>
#include <hip/hip_bf16.h>

typedef float v2f __attribute__((ext_vector_type(2)));
typedef float v8f __attribute__((ext_vector_type(8)));
typedef unsigned int u32x4 __attribute__((ext_vector_type(4)));
typedef int i32x4 __attribute__((ext_vector_type(4)));
typedef int i32x8 __attribute__((ext_vector_type(8)));

#define EPSF 1e-5f

// D = A(16x4) * B(4x16) + C, f32 WMMA.
// A layout: lane l (m=l&15, kh=l>>4): a.x=A[m][2kh], a.y=A[m][2kh+1]
// B layout: lane l (n=l&15, kh=l>>4): b.x=B[2kh][n], b.y=B[2kh+1][n]
// C/D layout: vgpr v, lane l: row = v + 8*(l>>4), col = l&15
static __device__ inline v8f wmma_f32(v2f a, v2f b, v8f c) {
  return __builtin_amdgcn_wmma_f32_16x16x4_f32(false, a, false, b, (short)0, c,
                                               false, false);
}

// ---------------------------------------------------------------------------
// Tensor Data Mover: linear DMA of nelem f32 (nelem <= 65535) global -> LDS.
// D# per CDNA5 ISA ch.8: group0 {count=1, lds_addr, global_addr, type=2},
// group1 {data_size=4B, tensor_dim0=nelem, tensor_dim1=1, tile_dim0=nelem,
//         tile_dim1=1, dim0_stride=nelem}. Groups 2/3 zero (2-D tensor).
// ---------------------------------------------------------------------------
static __device__ inline void tdm_copy_to_lds(const float* gsrc, float* ldst,
                                              unsigned nelem) {
  unsigned long long ga = (unsigned long long)(size_t)gsrc;
  unsigned lp = (unsigned)(size_t)(void*)ldst;  // low 32 bits = LDS offset
  u32x4 g0;
  g0[0] = 1u;                                   // count=1 (valid user D#)
  g0[1] = lp;                                   // lds_addr
  g0[2] = (unsigned)(ga & 0xffffffffu);         // global_addr[31:0]
  g0[3] = (unsigned)((ga >> 32) & 0x01ffffffu) | (2u << 30);  // addr[56:32]|type=2
  i32x8 g1;
  g1[0] = (int)(2u << 16);                      // data_size = 4 bytes
  g1[1] = (int)((nelem & 0xffffu) << 16);       // tensor_dim0[15:0]
  g1[2] = (int)((nelem >> 16) & 0xffffu) | (1 << 16);  // dim0[31:16] | tensor_dim1=1
  g1[3] = (int)((nelem & 0xffffu) << 16);       // tile_dim0 = nelem
  g1[4] = 1;                                    // tile_dim1 = 1, tile_dim2 = 0
  g1[5] = (int)nelem;                           // tensor_dim0_stride[31:0]
  g1[6] = 0;
  g1[7] = 0;
  i32x4 z4 = {0, 0, 0, 0};
  i32x8 z8 = {0, 0, 0, 0, 0, 0, 0, 0};
  __builtin_amdgcn_tensor_load_to_lds(g0, g1, z4, z4, z8, 0);
}

// ---------------------------------------------------------------------------
// small helpers
// ---------------------------------------------------------------------------
__global__ void zero_f32(float* __restrict__ p, int n) {
  int i = blockIdx.x * blockDim.x + threadIdx.x;
  if (i < n) p[i] = 0.f;
}

// pos_s[n] = {pos/sf[batch], reflectance}  (RYN gate == 1 analytically)
__global__ void build_pos_s(const float* __restrict__ pos, const int* __restrict__ batch,
                            const float* __restrict__ refl, const float* __restrict__ sf,
                            float* __restrict__ pos_s, int N) {
  int n = blockIdx.x * blockDim.x + threadIdx.x;
  if (n >= N) return;
  float s = sf[batch[n]];
  pos_s[n * 4 + 0] = pos[n * 3 + 0] / s;
  pos_s[n * 4 + 1] = pos[n * 3 + 1] / s;
  pos_s[n * 4 + 2] = pos[n * 3 + 2] / s;
  pos_s[n * 4 + 3] = refl[n];
}

// a = g * rsqrt(var + eps); c = be - mean * a   (BN as per-channel affine)
__global__ void finalize_bn(const float* __restrict__ sum, const float* __restrict__ sq,
                            float cnt, const float* __restrict__ g,
                            const float* __restrict__ be, float* __restrict__ A,
                            float* __restrict__ C, int n) {
  int i = blockIdx.x * blockDim.x + threadIdx.x;
  if (i >= n) return;
  float mean = sum[i] / cnt;
  float var = fmaxf(sq[i] / cnt - mean * mean, 0.f);
  float a = g[i] * rsqrtf(var + EPSF);
  A[i] = a;
  C[i] = be[i] - mean * a;
}

// BN(e*dw + db) folded: stats of a per-channel affine are analytic; db cancels.
// A = dw*g*rsqrt(dw^2*var_e + eps); C = be - mean_e * A
__global__ void prep_dw(const float* __restrict__ sum, const float* __restrict__ sq,
                        float cnt, const float* __restrict__ dw, const float* __restrict__ g,
                        const float* __restrict__ be, float* __restrict__ A,
                        float* __restrict__ C, int n) {
  int i = blockIdx.x * blockDim.x + threadIdx.x;
  if (i >= n) return;
  float mean = sum[i] / cnt;
  float var = fmaxf(sq[i] / cnt - mean * mean, 0.f);
  float d = dw[i];
  float a = d * g[i] * rsqrtf(d * d * var + EPSF);
  A[i] = a;
  C[i] = be[i] - mean * a;
}

// y = [relu]( x*A[c] + C[c] [+ res] ); optional per-channel stats of y.
// blockDim.x == Cch; one column per thread, rows strided by gridDim.
__global__ void elem_affine(const float* __restrict__ X, float* __restrict__ Y,
                            const float* __restrict__ A, const float* __restrict__ C,
                            const float* __restrict__ res, float* __restrict__ sum_o,
                            float* __restrict__ sq_o, int Mr, int Cch, int relu) {
  const int c = threadIdx.x;
  const float a = A[c], cc = C[c];
  float s = 0.f, q = 0.f;
  for (int r = blockIdx.x; r < Mr; r += gridDim.x) {
    size_t o = (size_t)r * Cch + c;
    float v = X[o] * a + cc;
    if (res) v += res[o];
    if (relu) v = fmaxf(v, 0.f);
    Y[o] = v;
    s += v;
    q += v * v;
  }
  if (sum_o) {
    atomicAdd(&sum_o[c], s);
    atomicAdd(&sq_o[c], q);
  }
}

// ---------------------------------------------------------------------------
// WMMA GEMM: Y[Mr,Nout] = X[Mr,Kin] @ W[Kin,Nout] + bias, + column stats.
// One wave = 32(M) x 64(N): two 16-row tiles share each B fragment (halves
// LDS traffic per WMMA). 8 waves/block. W staged through LDS in KC-row slabs,
// double-buffered via the Tensor Data Mover. A-operand loads are software-
// pipelined one K-step ahead to cover global latency.
// ---------------------------------------------------------------------------
#define GEMM_KC 32

__global__ __launch_bounds__(256) void gemm_bias_stats(
    const float* __restrict__ X, const float* __restrict__ W,
    const float* __restrict__ bias, float* __restrict__ Y, float* __restrict__ sum_o,
    float* __restrict__ sq_o, int Mr, int Kin, int Nout) {
  extern __shared__ float lds[];
  float* slab0 = lds;
  float* slab1 = lds + GEMM_KC * Nout;

  const int w = threadIdx.x >> 5, lane = threadIdx.x & 31;
  const int m = lane & 15, kh = lane >> 4;
  const int nt64 = Nout >> 6;
  const int gw = blockIdx.x * 8 + w;
  const int mt = gw / nt64;            // 32-row tile index
  const int nt = gw - mt * nt64;
  const bool active = (mt * 32 < Mr);  // wave-uniform; EXEC all-ones inside

  const float* xrow0 = X + (size_t)(mt * 32 + m) * Kin;
  const float* xrow1 = xrow0 + (size_t)16 * Kin;
  const int nslab = Kin / GEMM_KC;
  const unsigned slabElems = (unsigned)(GEMM_KC * Nout);

  if (w == 0) tdm_copy_to_lds(W, slab0, slabElems);

  v8f acc0[4] = {}, acc1[4] = {};
  v2f a0n = {}, a1n = {};
  if (active) {  // prime the A pipeline (K-step 0)
    a0n = *(const v2f*)(xrow0 + 2 * kh);
    a1n = *(const v2f*)(xrow1 + 2 * kh);
  }

  for (int s = 0; s < nslab; ++s) {
    float* cur = (s & 1) ? slab1 : slab0;
    float* nxt = (s & 1) ? slab0 : slab1;
    __syncthreads();  // everyone done reading the buffer we are about to refill
    if (w == 0) {
      if (s + 1 < nslab) {
        tdm_copy_to_lds(W + (size_t)(s + 1) * GEMM_KC * Nout, nxt, slabElems);
        __builtin_amdgcn_s_wait_tensorcnt(1);  // slab s landed (in-order TDM)
      } else {
        __builtin_amdgcn_s_wait_tensorcnt(0);
      }
    }
    __syncthreads();  // slab s visible to all waves
    if (active) {
      const int kbase = s * GEMM_KC;
      for (int kr = 0; kr < GEMM_KC; kr += 4) {
        v2f a0 = a0n, a1 = a1n;
        const int knext = kbase + kr + 4 + 2 * kh;
        if (knext < Kin) {  // software pipeline: next K-step's A loads now
          a0n = *(const v2f*)(xrow0 + knext);
          a1n = *(const v2f*)(xrow1 + knext);
          __builtin_prefetch(xrow0 + knext + 124, 0, 3);
          __builtin_prefetch(xrow1 + knext + 124, 0, 3);
        }
        const float* wsrc = cur + (size_t)(kr + 2 * kh) * Nout + nt * 64 + m;
#pragma unroll
        for (int t = 0; t < 4; ++t) {
          v2f b;
          b.x = wsrc[t * 16];
          b.y = wsrc[Nout + t * 16];
          acc0[t] = wmma_f32(a0, b, acc0[t]);
          acc1[t] = wmma_f32(a1, b, acc1[t]);
        }
      }
    }
  }

  if (!active) return;
#pragma unroll
  for (int t = 0; t < 4; ++t) {
    const int col = nt * 64 + t * 16 + m;
    const float bv = bias[col];
    float s = 0.f, q = 0.f;
#pragma unroll
    for (int v = 0; v < 8; ++v) {
      float val = acc0[t][v] + bv;
      Y[(size_t)(mt * 32 + v + 8 * kh) * Nout + col] = val;
      s += val;
      q += val * val;
    }
#pragma unroll
    for (int v = 0; v < 8; ++v) {
      float val = acc1[t][v] + bv;
      Y[(size_t)(mt * 32 + 16 + v + 8 * kh) * Nout + col] = val;
      s += val;
      q += val * val;
    }
    s += __shfl_xor(s, 16, 32);
    q += __shfl_xor(q, 16, 32);
    if (kh == 0) {
      atomicAdd(&sum_o[col], s);
      atomicAdd(&sq_o[col], q);
    }
  }
}

// ---------------------------------------------------------------------------
// Fused per-center local_nn: msg[16x36] -> 64 -> 64 -> 128 (-> rowmax).
// One wave per center (its 16 edges are contiguous). All weights TDM'd to LDS.
// PHASE 0: stats of relu(h2pre); PHASE 1: + stats of relu(h3pre);
// PHASE 2: normalize h3, per-center max -> agg[M,128].
// LDS layout (floats): stage 8192 | W1 2304 | W2 4096 | W3 8192 | lsum 128 | lsq 128
// ---------------------------------------------------------------------------
#define EL_STAGE 0
#define EL_W1 8192
#define EL_W2 (EL_W1 + 2304)
#define EL_W3 (EL_W2 + 4096)
#define EL_SUM (EL_W3 + 8192)
#define EL_SQ (EL_SUM + 128)
#define EL_TOTAL (EL_SQ + 128)

template <int PHASE>
__global__ __launch_bounds__(256) void edge_fused(
    const float* __restrict__ x, const float* __restrict__ pos_s,
    const int* __restrict__ edge_src, const int* __restrict__ idx,
    const float* __restrict__ W1, const float* __restrict__ b1,
    const float* __restrict__ W2, const float* __restrict__ b2,
    const float* __restrict__ W3, const float* __restrict__ b3,
    const float* __restrict__ a2, const float* __restrict__ c2,
    const float* __restrict__ a3, const float* __restrict__ c3,
    float* __restrict__ sum_o, float* __restrict__ sq_o, float* __restrict__ agg) {
  extern __shared__ float lds[];
  float* w1l = lds + EL_W1;
  float* w2l = lds + EL_W2;
  float* w3l = lds + EL_W3;
  float* lsum = lds + EL_SUM;
  float* lsq = lds + EL_SQ;

  const int tid = threadIdx.x;
  const int w = tid >> 5, lane = tid & 31;
  const int m = lane & 15, kh = lane >> 4;
  float* st = lds + EL_STAGE + w * 1024;  // per-wave 16x64 transpose buffer

  if (tid < 128) {
    lsum[tid] = 0.f;
    lsq[tid] = 0.f;
  }
  if (w == 0) {
    tdm_copy_to_lds(W1, w1l, 2304);
    tdm_copy_to_lds(W2, w2l, 4096);
    tdm_copy_to_lds(W3, w3l, 8192);
    __builtin_amdgcn_s_wait_tensorcnt(0);
  }
  __syncthreads();

  const int j = blockIdx.x * 8 + w;  // center index (grid sized exactly M/8)
  const int e = j * 16 + m;          // this lane's edge row
  const int src = edge_src[e];
  const int ci = idx[j];
  v2f prel;  // rel pair for K-step 32: kh=0 -> comps {0,1}, kh=1 -> {2,3}
  prel.x = pos_s[src * 4 + 2 * kh] - pos_s[ci * 4 + 2 * kh];
  prel.y = pos_s[src * 4 + 2 * kh + 1] - pos_s[ci * 4 + 2 * kh + 1];
  const float* xr = x + (size_t)src * 32;
  __builtin_prefetch(xr, 0, 3);

  // ---- layer 1: msg[16x36] @ W1[36x64], relu ----
  v8f h1[4] = {};
#pragma unroll
  for (int k0 = 0; k0 <= 32; k0 += 4) {
    v2f a = (k0 < 32) ? *(const v2f*)(xr + k0 + 2 * kh) : prel;
    const float* w0 = w1l + (k0 + 2 * kh) * 64 + m;
#pragma unroll
    for (int t = 0; t < 4; ++t) {
      v2f b;
      b.x = w0[t * 16];
      b.y = w0[64 + t * 16];
      h1[t] = wmma_f32(a, b, h1[t]);
    }
  }
#pragma unroll
  for (int t = 0; t < 4; ++t) {
    const float bv = b1[t * 16 + m];
#pragma unroll
    for (int v = 0; v < 8; ++v)
      st[(v + 8 * kh) * 64 + t * 16 + m] = fmaxf(h1[t][v] + bv, 0.f);
  }
  __syncthreads();

  // ---- layer 2: h1[16x64] @ W2[64x64], relu ----
  v8f h2[4] = {};
#pragma unroll
  for (int k0 = 0; k0 < 64; k0 += 4) {
    v2f a = *(const v2f*)(&st[m * 64 + k0 + 2 * kh]);
    const float* w0 = w2l + (k0 + 2 * kh) * 64 + m;
#pragma unroll
    for (int t = 0; t < 4; ++t) {
      v2f b;
      b.x = w0[t * 16];
      b.y = w0[64 + t * 16];
      h2[t] = wmma_f32(a, b, h2[t]);
    }
  }
#pragma unroll
  for (int t = 0; t < 4; ++t) {
    const float bv = b2[t * 16 + m];
#pragma unroll
    for (int v = 0; v < 8; ++v) h2[t][v] = fmaxf(h2[t][v] + bv, 0.f);
  }

  if (PHASE == 0) {  // BN2 stats over relu(h2pre)
#pragma unroll
    for (int t = 0; t < 4; ++t) {
      float s = 0.f, q = 0.f;
#pragma unroll
      for (int v = 0; v < 8; ++v) {
        s += h2[t][v];
        q += h2[t][v] * h2[t][v];
      }
      s += __shfl_xor(s, 16, 32);
      q += __shfl_xor(q, 16, 32);
      if (kh == 0) {
        atomicAdd(&lsum[t * 16 + m], s);
        atomicAdd(&lsq[t * 16 + m], q);
      }
    }
    __syncthreads();
    if (tid < 64) {
      atomicAdd(&sum_o[tid], lsum[tid]);
      atomicAdd(&sq_o[tid], lsq[tid]);
    }
    return;
  }

  // normalize h2 with BN2 affine and restage in A-layout
  __syncthreads();
#pragma unroll
  for (int t = 0; t < 4; ++t) {
    const float aa = a2[t * 16 + m], cc = c2[t * 16 + m];
#pragma unroll
    for (int v = 0; v < 8; ++v)
      st[(v + 8 * kh) * 64 + t * 16 + m] = h2[t][v] * aa + cc;
  }
  __syncthreads();

  // ---- layer 3: h2n[16x64] @ W3[64x128], relu ----
  v8f h3[8] = {};
#pragma unroll
  for (int k0 = 0; k0 < 64; k0 += 4) {
    v2f a = *(const v2f*)(&st[m * 64 + k0 + 2 * kh]);
    const float* w0 = w3l + (k0 + 2 * kh) * 128 + m;
#pragma unroll
    for (int t = 0; t < 8; ++t) {
      v2f b;
      b.x = w0[t * 16];
      b.y = w0[128 + t * 16];
      h3[t] = wmma_f32(a, b, h3[t]);
    }
  }
#pragma unroll
  for (int t = 0; t < 8; ++t) {
    const float bv = b3[t * 16 + m];
#pragma unroll
    for (int v = 0; v < 8; ++v) h3[t][v] = fmaxf(h3[t][v] + bv, 0.f);
  }

  if (PHASE == 1) {  // BN3 stats over relu(h3pre)
#pragma unroll
    for (int t = 0; t < 8; ++t) {
      float s = 0.f, q = 0.f;
#pragma unroll
      for (int v = 0; v < 8; ++v) {
        s += h3[t][v];
        q += h3[t][v] * h3[t][v];
      }
      s += __shfl_xor(s, 16, 32);
      q += __shfl_xor(q, 16, 32);
      if (kh == 0) {
        atomicAdd(&lsum[t * 16 + m], s);
        atomicAdd(&lsq[t * 16 + m], q);
      }
    }
    __syncthreads();
    if (tid < 128) {
      atomicAdd(&sum_o[tid], lsum[tid]);
      atomicAdd(&sq_o[tid], lsq[tid]);
    }
    return;
  }

  // PHASE 2: normalize with BN3 and max over this center's 16 edges
#pragma unroll
  for (int t = 0; t < 8; ++t) {
    const int ch = t * 16 + m;
    const float aa = a3[ch], cc = c3[ch];
    float mx = -3.402823466e+38f;
#pragma unroll
    for (int v = 0; v < 8; ++v) mx = fmaxf(mx, h3[t][v] * aa + cc);
    mx = fmaxf(mx, __shfl_xor(mx, 16, 32));
    if (kh == 0) agg[(size_t)j * 128 + ch] = mx;
  }
}

// tail outputs: pos[idx], float(batch[idx]), reflectance[idx], sf
__global__ void tail_outputs(const float* __restrict__ pos, const int* __restrict__ batch,
                             const float* __restrict__ refl, const float* __restrict__ sf,
                             const int* __restrict__ idx, float* __restrict__ o_pos,
                             float* __restrict__ o_batch, float* __restrict__ o_refl,
                             float* __restrict__ o_sf, int Mr, int Bn) {
  int i = blockIdx.x * blockDim.x + threadIdx.x;
  if (i < Mr) {
    int j = idx[i];
    o_pos[i * 3 + 0] = pos[j * 3 + 0];
    o_pos[i * 3 + 1] = pos[j * 3 + 1];
    o_pos[i * 3 + 2] = pos[j * 3 + 2];
    o_batch[i] = (float)batch[j];
    o_refl[i] = refl[j];
  }
  if (i < Bn) o_sf[i] = sf[i];
}

// ---------------------------------------------------------------------------
extern "C" void kernel_launch(void* const* d_in, const int* in_sizes, int n_in,
                              void* d_out, int out_size, void* d_ws, size_t ws_size,
                              hipStream_t stream) {
  const float* x = (const float*)d_in[0];
  const float* pos = (const float*)d_in[1];
  const int* batch = (const int*)d_in[2];
  const float* refl = (const float*)d_in[3];
  const float* sf = (const float*)d_in[4];
  const int* idx = (const int*)d_in[5];
  const int* esrc = (const int*)d_in[6];
  // d_in[7] edge_dst: structurally repeat(arange(M),16) -> implicit
  const float* W1 = (const float*)d_in[8];
  const float* b1 = (const float*)d_in[9];
  const float* W2 = (const float*)d_in[10];
  const float* b2 = (const float*)d_in[11];
  const float* g2 = (const float*)d_in[12];
  const float* be2 = (const float*)d_in[13];
  const float* W3 = (const float*)d_in[14];
  const float* b3 = (const float*)d_in[15];
  const float* g3 = (const float*)d_in[16];
  const float* be3 = (const float*)d_in[17];
  // d_in[18..23] RYN weights: gate == 1 analytically, unused
  const float* expW = (const float*)d_in[24];
  const float* expB = (const float*)d_in[25];
  const float* expG = (const float*)d_in[26];
  const float* expBe = (const float*)d_in[27];
  const float* dwW = (const float*)d_in[28];  // [2,512]; dw_b cancels analytically
  const float* dwG = (const float*)d_in[30];
  const float* dwBe = (const float*)d_in[31];
  const float* pwW = (const float*)d_in[32];  // [2,512,512]
  const float* pwB = (const float*)d_in[33];
  const float* pwG = (const float*)d_in[34];
  const float* pwBe = (const float*)d_in[35];
  const float* cG = (const float*)d_in[36];
  const float* cBe = (const float*)d_in[37];
  const float* prW = (const float*)d_in[38];
  const float* prB = (const float*)d_in[39];
  const float* prG = (const float*)d_in[40];
  const float* prBe = (const float*)d_in[41];

  const int N = in_sizes[3];
  const int Bn = in_sizes[4];
  const int M = in_sizes[5];
  const int E = in_sizes[6];

  float* ws = (float*)d_ws;
  float* pos_s = ws;                    // N*4
  float* agg = pos_s + (size_t)N * 4;   // M*128
  float* bufA = agg + (size_t)M * 128;  // M*512
  float* bufB = bufA + (size_t)M * 512; // M*512
  float* st = bufB + (size_t)M * 512;   // stats/coeffs (2816 floats)
  float* sum2 = st + 0;
  float* sq2 = st + 64;
  float* a2 = st + 128;
  float* c2 = st + 192;
  float* sum3 = st + 256;
  float* sq3 = st + 384;
  float* a3 = st + 512;
  float* c3 = st + 640;
  float* sumS = st + 768;  // sumS+sqS contiguous 1024 floats for re-zeroing
  float* sqS = st + 1280;
  float* aS = st + 1792;
  float* cS = st + 2304;

  float* outp = (float*)d_out;
  float* oMain = outp;                     // [M,128]
  float* oPos = outp + (size_t)M * 128;    // [M,3]
  float* oBatch = outp + (size_t)M * 131;  // [M]
  float* oRefl = outp + (size_t)M * 132;   // [M]
  float* oSf = outp + (size_t)M * 133;     // [B]

  const float fE = (float)E;
  const float fM = (float)M;
  const size_t edgeSmem = (size_t)EL_TOTAL * sizeof(float);
  const size_t g512Smem = (size_t)2 * GEMM_KC * 512 * sizeof(float);
  const size_t g128Smem = (size_t)2 * GEMM_KC * 128 * sizeof(float);

  zero_f32<<<(2816 + 255) / 256, 256, 0, stream>>>(st, 2816);
  build_pos_s<<<(N + 255) / 256, 256, 0, stream>>>(pos, batch, refl, sf, pos_s, N);

  // ---- local_nn: 3 sweeps (recompute; x/pos_s stay L2-resident) ----
  dim3 eg(M / 8);
  edge_fused<0><<<eg, 256, edgeSmem, stream>>>(x, pos_s, esrc, idx, W1, b1, W2, b2,
                                               W3, b3, a2, c2, a3, c3, sum2, sq2, agg);
  finalize_bn<<<1, 64, 0, stream>>>(sum2, sq2, fE, g2, be2, a2, c2, 64);
  edge_fused<1><<<eg, 256, edgeSmem, stream>>>(x, pos_s, esrc, idx, W1, b1, W2, b2,
                                               W3, b3, a2, c2, a3, c3, sum3, sq3, agg);
  finalize_bn<<<1, 128, 0, stream>>>(sum3, sq3, fE, g3, be3, a3, c3, 128);
  edge_fused<2><<<eg, 256, edgeSmem, stream>>>(x, pos_s, esrc, idx, W1, b1, W2, b2,
                                               W3, b3, a2, c2, a3, c3, sum3, sq3, agg);

  // ---- InvertedResidualBlock ----
  auto gblocks = [](int Mr, int Nout) { return ((Mr / 32) * (Nout / 64) + 7) / 8; };

  // expand: t1 = agg @ expW + expB ; stats -> BN affine
  gemm_bias_stats<<<gblocks(M, 512), 256, g512Smem, stream>>>(agg, expW, expB, bufA,
                                                              sumS, sqS, M, 128, 512);
  finalize_bn<<<2, 256, 0, stream>>>(sumS, sqS, fM, expG, expBe, aS, cS, 512);
  zero_f32<<<4, 256, 0, stream>>>(sumS, 1024);
  // e = relu(bn(t1)); stats of e feed analytic depthwise-BN fold
  elem_affine<<<480, 512, 0, stream>>>(bufA, bufB, aS, cS, nullptr, sumS, sqS, M, 512, 1);
  prep_dw<<<2, 256, 0, stream>>>(sumS, sqS, fM, dwW, dwG, dwBe, aS, cS, 512);
  elem_affine<<<480, 512, 0, stream>>>(bufB, bufA, aS, cS, nullptr, nullptr, nullptr, M, 512, 1);  // d0
  zero_f32<<<4, 256, 0, stream>>>(sumS, 1024);
  gemm_bias_stats<<<gblocks(M, 512), 256, g512Smem, stream>>>(bufA, pwW, pwB, bufB,
                                                              sumS, sqS, M, 512, 512);
  finalize_bn<<<2, 256, 0, stream>>>(sumS, sqS, fM, pwG, pwBe, aS, cS, 512);
  zero_f32<<<4, 256, 0, stream>>>(sumS, 1024);
  elem_affine<<<480, 512, 0, stream>>>(bufB, bufA, aS, cS, nullptr, sumS, sqS, M, 512, 1);  // p0
  finalize_bn<<<2, 256, 0, stream>>>(sumS, sqS, fM, cG, cBe, aS, cS, 512);
  zero_f32<<<4, 256, 0, stream>>>(sumS, 1024);
  elem_affine<<<480, 512, 0, stream>>>(bufA, bufB, aS, cS, nullptr, sumS, sqS, M, 512, 1);  // cmid0
  prep_dw<<<2, 256, 0, stream>>>(sumS, sqS, fM, dwW + 512, dwG + 512, dwBe + 512, aS, cS, 512);
  elem_affine<<<480, 512, 0, stream>>>(bufB, bufA, aS, cS, nullptr, nullptr, nullptr, M, 512, 1);  // d1
  zero_f32<<<4, 256, 0, stream>>>(sumS, 1024);
  gemm_bias_stats<<<gblocks(M, 512), 256, g512Smem, stream>>>(
      bufA, pwW + (size_t)512 * 512, pwB + 512, bufB, sumS, sqS, M, 512, 512);
  finalize_bn<<<2, 256, 0, stream>>>(sumS, sqS, fM, pwG + 512, pwBe + 512, aS, cS, 512);
  zero_f32<<<4, 256, 0, stream>>>(sumS, 1024);
  elem_affine<<<480, 512, 0, stream>>>(bufB, bufA, aS, cS, nullptr, sumS, sqS, M, 512, 1);  // p1
  finalize_bn<<<2, 256, 0, stream>>>(sumS, sqS, fM, cG + 512, cBe + 512, aS, cS, 512);
  elem_affine<<<480, 512, 0, stream>>>(bufA, bufB, aS, cS, nullptr, nullptr, nullptr, M, 512, 0);  // cmid1 (no relu)
  zero_f32<<<4, 256, 0, stream>>>(sumS, 1024);
  // project: opre = cmid1 @ prW + prB
  gemm_bias_stats<<<gblocks(M, 128), 256, g128Smem, stream>>>(bufB, prW, prB, bufA,
                                                              sumS, sqS, M, 512, 128);
  finalize_bn<<<1, 128, 0, stream>>>(sumS, sqS, fM, prG, prBe, aS, cS, 128);
  // out = relu(bn(opre) + agg)
  elem_affine<<<480, 128, 0, stream>>>(bufA, oMain, aS, cS, agg, nullptr, nullptr, M, 128, 1);

  tail_outputs<<<(M + 255) / 256, 256, 0, stream>>>(pos, batch, refl, sf, idx, oPos,
                                                    oBatch, oRefl, oSf, M, Bn);
}